// GINEVirtualNodeEncoder_39685497815719
// MI455X (gfx1250) — compile-verified
//
#include <hip/hip_runtime.h>
#include <hip/hip_bf16.h>

typedef __attribute__((ext_vector_type(2))) float v2f;
typedef __attribute__((ext_vector_type(8))) float v8f;

#define H_DIM   128
#define NGRAPH  512
#define NLAYERS 3
#define BN_EPS  1e-5f
#define BN_ROWS 256

// ---------------------------------------------------------------- zero
__global__ void k_zero(float* __restrict__ p, int n) {
  int i = blockIdx.x * blockDim.x + threadIdx.x;
  if (i < n) p[i] = 0.0f;
}

// ---------------------------------------------------------------- counts
__global__ void k_counts(const int* __restrict__ batch, float* __restrict__ counts, int n) {
  int i = blockIdx.x * blockDim.x + threadIdx.x;
  if (i < n) atomicAdd(&counts[batch[i]], 1.0f);
}

// ---------------------------------------------------------------- WMMA GEMM
// C[M,Nc] = act(A[M,K] @ W[K,Nc] + bias[Nc])   (optionally C += result)
// One wave per 16x16 tile; fp32 WMMA 16x16x4.
// A frag (16x4 fp32): lanes 0-15 hold K+0,K+1 ; lanes 16-31 hold K+2,K+3.
// B frag (4x16 fp32): mirrored; C/D: 8 VGPRs, lanes 0-15 rows 0-7, 16-31 rows 8-15.
template<bool RELU, bool ACCUM>
__global__ void k_gemm_wmma(const float* __restrict__ A,
                            const float* __restrict__ W,
                            const float* __restrict__ bias,
                            float* __restrict__ C,
                            int M, int K, int Nc) {
  const int lane          = threadIdx.x & 31;
  const int wave          = threadIdx.x >> 5;
  const int wavesPerBlock = blockDim.x >> 5;
  const int tile          = blockIdx.x * wavesPerBlock + wave;
  const int tilesN        = Nc >> 4;
  const int tilesTotal    = (M >> 4) * tilesN;
  if (tile >= tilesTotal) return;          // wave-uniform: EXEC all-1 for WMMA
  const int row0 = (tile / tilesN) << 4;
  const int col0 = (tile % tilesN) << 4;

  const int mrow = row0 + (lane & 15);     // A row supplied by this lane
  const int ncol = col0 + (lane & 15);     // B/C column held by this lane
  const int kh   = (lane >> 4) << 1;       // upper half-wave holds K+2,K+3

  v8f acc = {0.f, 0.f, 0.f, 0.f, 0.f, 0.f, 0.f, 0.f};
  for (int k = 0; k < K; k += 4) {
    v2f a, b;
    const float* ap = A + (size_t)mrow * K + (k + kh);
    a[0] = ap[0];
    a[1] = ap[1];
    const float* wp = W + (size_t)(k + kh) * Nc + ncol;
    b[0] = wp[0];
    b[1] = wp[Nc];
    acc = __builtin_amdgcn_wmma_f32_16x16x4_f32(false, a, false, b,
                                                (short)0, acc, false, false);
  }

  const float bv    = bias[ncol];
  const int   rbase = row0 + ((lane >> 4) << 3);
#pragma unroll
  for (int v = 0; v < 8; ++v) {
    float val = acc[v] + bv;
    if (RELU) val = fmaxf(val, 0.0f);
    float* outp = C + (size_t)(rbase + v) * Nc + ncol;
    if (ACCUM) *outp += val; else *outp = val;
  }
}

// ---------------------------------------------------------------- vn broadcast
// h[n] += vn[batch[n]] ; z[n] = h[n]   (one wave per node, float4 per lane)
__global__ void k_vn_broadcast(float* __restrict__ h, float* __restrict__ z,
                               const float* __restrict__ vn,
                               const int* __restrict__ batch, int n) {
  int gid = blockIdx.x * blockDim.x + threadIdx.x;
  int nd  = gid >> 5;
  if (nd >= n) return;
  int c0 = (gid & 31) << 2;
  int g  = batch[nd];
  const float4 vv = *(const float4*)(vn + (size_t)g  * H_DIM + c0);
  float4       hv = *(const float4*)(h  + (size_t)nd * H_DIM + c0);
  hv.x += vv.x; hv.y += vv.y; hv.z += vv.z; hv.w += vv.w;
  *(float4*)(h + (size_t)nd * H_DIM + c0) = hv;
  *(float4*)(z + (size_t)nd * H_DIM + c0) = hv;
}

// ---------------------------------------------------------------- edge messages
// z[dst] += relu(h[src] + edge_attr@edge_w + edge_b), edge MLP fused (K=2).
// One wave per edge (grid-stride); edge-weight columns live in registers.
__global__ void k_msg_agg(const float* __restrict__ h,  const float* __restrict__ ea,
                          const float* __restrict__ ew, const float* __restrict__ eb,
                          const int* __restrict__ src,  const int* __restrict__ dst,
                          float* __restrict__ z, int ecnt) {
  int lane   = threadIdx.x & 31;
  int gwave  = (blockIdx.x * blockDim.x + threadIdx.x) >> 5;
  int nwaves = (gridDim.x * blockDim.x) >> 5;
  int c0     = lane << 2;
  const float4 w0 = *(const float4*)(ew + c0);
  const float4 w1 = *(const float4*)(ew + H_DIM + c0);
  const float4 bb = *(const float4*)(eb + c0);
  for (int e = gwave; e < ecnt; e += nwaves) {
    int   s  = src[e];
    int   d  = dst[e];
    float a0 = ea[2 * e + 0];
    float a1 = ea[2 * e + 1];
    const float4 hv = *(const float4*)(h + (size_t)s * H_DIM + c0);
    float m0 = fmaxf(hv.x + a0 * w0.x + a1 * w1.x + bb.x, 0.0f);
    float m1 = fmaxf(hv.y + a0 * w0.y + a1 * w1.y + bb.y, 0.0f);
    float m2 = fmaxf(hv.z + a0 * w0.z + a1 * w1.z + bb.z, 0.0f);
    float m3 = fmaxf(hv.w + a0 * w0.w + a1 * w1.w + bb.w, 0.0f);
    float* zp = z + (size_t)d * H_DIM + c0;
    atomicAdd(zp + 0, m0);
    atomicAdd(zp + 1, m1);
    atomicAdd(zp + 2, m2);
    atomicAdd(zp + 3, m3);
  }
}

// ---------------------------------------------------------------- BN stats
// stats[c] += sum(z[:,c]); stats[H+c] += sum(z[:,c]^2) over a row chunk.
__global__ void k_bn_stats(const float* __restrict__ z, float* __restrict__ stats, int n) {
  int c  = threadIdx.x;                 // 0..127, coalesced across the row
  int r0 = blockIdx.x * BN_ROWS;
  int r1 = r0 + BN_ROWS; if (r1 > n) r1 = n;
  float s = 0.0f, s2 = 0.0f;
  for (int r = r0; r < r1; ++r) {
    float v = z[(size_t)r * H_DIM + c];
    s  += v;
    s2 += v * v;
  }
  atomicAdd(&stats[c], s);
  atomicAdd(&stats[H_DIM + c], s2);
}

// ---------------------------------------------------------------- BN apply + relu + pool
__global__ void k_bn_apply_pool(const float* __restrict__ z, const float* __restrict__ stats,
                                const float* __restrict__ gamma, const float* __restrict__ beta,
                                const int* __restrict__ batch, float* __restrict__ h,
                                float* __restrict__ pooled, float invN, int n) {
  int gid = blockIdx.x * blockDim.x + threadIdx.x;
  int nd  = gid >> 5;
  if (nd >= n) return;
  int c0 = (gid & 31) << 2;
  int g  = batch[nd];
  float4 zv = *(const float4*)(z + (size_t)nd * H_DIM + c0);
  float in[4] = {zv.x, zv.y, zv.z, zv.w};
  float out[4];
#pragma unroll
  for (int j = 0; j < 4; ++j) {
    int   c   = c0 + j;
    float mu  = stats[c] * invN;
    float var = stats[H_DIM + c] * invN - mu * mu;
    float sc  = gamma[c] * rsqrtf(var + BN_EPS);
    float v   = (in[j] - mu) * sc + beta[c];
    v = fmaxf(v, 0.0f);
    out[j] = v;
    atomicAdd(&pooled[(size_t)g * H_DIM + c], v);
  }
  float4 hv = {out[0], out[1], out[2], out[3]};
  *(float4*)(h + (size_t)nd * H_DIM + c0) = hv;
}

// ---------------------------------------------------------------- graph embedding
__global__ void k_graph_emb(const float* __restrict__ pooled, const float* __restrict__ counts,
                            float* __restrict__ out, int n) {
  int i = blockIdx.x * blockDim.x + threadIdx.x;
  if (i < n) {
    float cnt = counts[i / H_DIM];
    out[i] = pooled[i] / fmaxf(cnt, 1.0f);
  }
}

// ================================================================ host
extern "C" void kernel_launch(void* const* d_in, const int* in_sizes, int n_in,
                              void* d_out, int out_size, void* d_ws, size_t ws_size,
                              hipStream_t stream) {
  (void)n_in; (void)out_size; (void)ws_size;

  const float* x      = (const float*)d_in[0];
  const float* eattr  = (const float*)d_in[1];
  const int*   eidx   = (const int*)  d_in[2];
  const int*   batch  = (const int*)  d_in[3];
  const float* proj_w = (const float*)d_in[4];
  const float* proj_b = (const float*)d_in[5];
  const float* edge_w = (const float*)d_in[6];
  const float* edge_b = (const float*)d_in[7];
  const float* cw1    = (const float*)d_in[8];
  const float* cb1    = (const float*)d_in[9];
  const float* cw2    = (const float*)d_in[10];
  const float* cb2    = (const float*)d_in[11];
  const float* bng    = (const float*)d_in[12];
  const float* bnb    = (const float*)d_in[13];
  const float* vnw1   = (const float*)d_in[14];
  const float* vnb1   = (const float*)d_in[15];
  const float* vnw2   = (const float*)d_in[16];
  const float* vnb2   = (const float*)d_in[17];

  const int N = in_sizes[0] / 64;   // IN_DIM = 64
  const int E = in_sizes[1] / 2;    // EA_DIM = 2
  const int* src = eidx;
  const int* dst = eidx + E;

  float* h    = (float*)d_out;                  // [N, H]  final node features
  float* gemb = h + (size_t)N * H_DIM;          // [NG, H] graph embeddings

  float* ws     = (float*)d_ws;
  float* z      = ws;  ws += (size_t)N * H_DIM; // agg/accumulator
  float* t      = ws;  ws += (size_t)N * H_DIM; // GEMM ping-pong
  float* vn     = ws;  ws += NGRAPH * H_DIM;
  float* pooled = ws;  ws += NGRAPH * H_DIM;
  float* tvn    = ws;  ws += NGRAPH * H_DIM;
  float* stats  = ws;  ws += 2 * H_DIM;
  float* counts = ws;  ws += NGRAPH;

  // one-time init (all accumulation targets re-zeroed every call -> replay-safe)
  k_zero<<<(NGRAPH * H_DIM + 255) / 256, 256, 0, stream>>>(vn, NGRAPH * H_DIM);
  k_zero<<<(NGRAPH + 255) / 256, 256, 0, stream>>>(counts, NGRAPH);
  k_counts<<<(N + 255) / 256, 256, 0, stream>>>(batch, counts, N);

  // h = x @ proj_w + proj_b
  {
    int tiles = (N / 16) * (H_DIM / 16);
    k_gemm_wmma<false, false><<<(tiles + 7) / 8, 256, 0, stream>>>(
        x, proj_w, proj_b, h, N, 64, H_DIM);
  }

  const int tilesNH = (N / 16) * (H_DIM / 16);
  const int tilesVN = (NGRAPH / 16) * (H_DIM / 16);
  const int nodeThreads = N * 32;               // one wave per node

  for (int i = 0; i < NLAYERS; ++i) {
    k_zero<<<(NGRAPH * H_DIM + 255) / 256, 256, 0, stream>>>(pooled, NGRAPH * H_DIM);
    k_zero<<<1, 256, 0, stream>>>(stats, 2 * H_DIM);

    // h += vn[batch]; z = h
    k_vn_broadcast<<<(nodeThreads + 255) / 256, 256, 0, stream>>>(h, z, vn, batch, N);

    // z[dst] += relu(h[src] + edge MLP)   (edge features computed on the fly)
    k_msg_agg<<<2048, 256, 0, stream>>>(h, eattr, edge_w, edge_b, src, dst, z, E);

    // conv MLP: t = relu(z@w1+b1); z = t@w2+b2
    k_gemm_wmma<true, false><<<(tilesNH + 7) / 8, 256, 0, stream>>>(
        z, cw1 + (size_t)i * H_DIM * H_DIM, cb1 + i * H_DIM, t, N, H_DIM, H_DIM);
    k_gemm_wmma<false, false><<<(tilesNH + 7) / 8, 256, 0, stream>>>(
        t, cw2 + (size_t)i * H_DIM * H_DIM, cb2 + i * H_DIM, z, N, H_DIM, H_DIM);

    // BatchNorm (training stats) + relu, fused with per-graph pooling
    k_bn_stats<<<(N + BN_ROWS - 1) / BN_ROWS, H_DIM, 0, stream>>>(z, stats, N);
    k_bn_apply_pool<<<(nodeThreads + 255) / 256, 256, 0, stream>>>(
        z, stats, bng + i * H_DIM, bnb + i * H_DIM, batch, h, pooled,
        1.0f / (float)N, N);

    // virtual-node MLP update
    if (i != NLAYERS - 1) {
      k_gemm_wmma<true, false><<<(tilesVN + 7) / 8, 256, 0, stream>>>(
          pooled, vnw1, vnb1, tvn, NGRAPH, H_DIM, H_DIM);
      k_gemm_wmma<false, true><<<(tilesVN + 7) / 8, 256, 0, stream>>>(
          tvn, vnw2, vnb2, vn, NGRAPH, H_DIM, H_DIM);
    }
  }

  // graph_emb = pooled(last layer) / max(counts, 1)
  k_graph_emb<<<(NGRAPH * H_DIM + 255) / 256, 256, 0, stream>>>(
      pooled, counts, gemb, NGRAPH * H_DIM);
}